// MultiScalePeriodicPatchEmbedding_57939108823201
// MI455X (gfx1250) — compile-verified
//
#include <hip/hip_runtime.h>
#include <hip/hip_bf16.h>
#include <math.h>

#define SEQ_LEN   96
#define NUM_VARS  14
#define FREQ      48
#define HID       192      // FREQ*HF
#define GOUT      96       // 2*FREQ == SEQ_LEN
#define D_MODEL   512
#define TOPK      5
#define N_EXPERTS 95
#define MAX_IN    1344     // SEQ_LEN*NUM_VARS
#define B_BATCH   512
#define LAM       0.01f

typedef float v2f __attribute__((ext_vector_type(2)));
typedef float v8f __attribute__((ext_vector_type(8)));
// matches the builtin's "V4i" prototype (generic vector kind)
typedef int b128_t __attribute__((vector_size(16)));
typedef __attribute__((address_space(1))) b128_t* glob128_t;
typedef __attribute__((address_space(3))) b128_t* lds128_t;

#if __has_builtin(__builtin_amdgcn_global_load_async_to_lds_b128) && \
    __has_builtin(__builtin_amdgcn_s_wait_asynccnt)
#define USE_ASYNC_LDS 1
#else
#define USE_ASYNC_LDS 0
#endif

// ---------------------------------------------------------------------------
// Kernel 1: gating. One block per batch row. 192 threads (== HID).
// DFT (drop DC) -> complex MLP 48->192 (relu) -> 192->96 -> softshrink ->
// magnitude, mean over 14 channels -> top-5 -> softmax -> gates row.
// ---------------------------------------------------------------------------
__global__ void __launch_bounds__(192)
gates_kernel(const float* __restrict__ x,
             const float* __restrict__ w1, const float* __restrict__ b1,
             const float* __restrict__ w2, const float* __restrict__ b2,
             float* __restrict__ gatesOut) {
  const int b   = blockIdx.x;
  const int tid = threadIdx.x;

  __shared__ float xloc[SEQ_LEN * NUM_VARS];
  __shared__ float fr[FREQ], fi[FREQ];
  __shared__ float hr[HID], hi[HID];
  __shared__ float wacc[GOUT];
  __shared__ float grow[GOUT];
  __shared__ float tv[TOPK];
  __shared__ int   tix[TOPK];

  for (int i = tid; i < SEQ_LEN * NUM_VARS; i += blockDim.x)
    xloc[i] = x[(size_t)b * SEQ_LEN * NUM_VARS + i];
  if (tid < GOUT) { wacc[tid] = 0.f; grow[tid] = 0.f; }
  __syncthreads();

  const float* w1r = w1;  const float* w1i = w1 + FREQ * HID;
  const float* b1r = b1;  const float* b1i = b1 + HID;
  const float* w2r = w2;  const float* w2i = w2 + HID * GOUT;
  const float* b2r = b2;  const float* b2i = b2 + GOUT;

  for (int c = 0; c < NUM_VARS; ++c) {
    // rFFT bins k = 1..48 (DC dropped).  X[k] = sum_t x[t] (cos - i sin)
    if (tid < FREQ) {
      const int   k   = tid + 1;
      const float ang = 6.28318530717958647692f * (float)k / (float)SEQ_LEN;
      float sr = 0.f, si = 0.f;
      for (int t = 0; t < SEQ_LEN; ++t) {
        float s, cs;
        __sincosf(ang * (float)t, &s, &cs);
        const float v = xloc[t * NUM_VARS + c];
        sr += v * cs;
        si -= v * s;
      }
      fr[tid] = sr; fi[tid] = si;
    }
    __syncthreads();
    // hidden layer (complex linear + relu on each part), 192 outputs
    {
      const int h = tid;
      float ar = b1r[h], ai = b1i[h];
      for (int f = 0; f < FREQ; ++f) {
        const float wr = w1r[f * HID + h], wi = w1i[f * HID + h];
        const float xr = fr[f], xim = fi[f];
        ar = fmaf(xr, wr, fmaf(-xim, wi, ar));
        ai = fmaf(xim, wr, fmaf(xr, wi, ai));
      }
      hr[h] = ar > 0.f ? ar : 0.f;
      hi[h] = ai > 0.f ? ai : 0.f;
    }
    __syncthreads();
    // output layer, softshrink, magnitude accumulate
    if (tid < GOUT) {
      const int g = tid;
      float orr = b2r[g], oi = b2i[g];
      for (int h2 = 0; h2 < HID; ++h2) {
        const float wr = w2r[h2 * GOUT + g], wi = w2i[h2 * GOUT + g];
        const float a = hr[h2], bb = hi[h2];
        orr = fmaf(a, wr, fmaf(-bb, wi, orr));
        oi  = fmaf(bb, wr, fmaf(a,  wi, oi));
      }
      orr = orr > LAM ? orr - LAM : (orr < -LAM ? orr + LAM : 0.f);
      oi  = oi  > LAM ? oi  - LAM : (oi  < -LAM ? oi  + LAM : 0.f);
      wacc[g] += sqrtf(orr * orr + oi * oi);
    }
    __syncthreads();
  }

  if (tid < GOUT) wacc[tid] *= (1.f / (float)NUM_VARS);
  __syncthreads();

  // top-5 (descending, ties -> lower index) + softmax + scatter
  if (tid == 0) {
    for (int j = 0; j < TOPK; ++j) {
      float best = -INFINITY; int bi = 0;
      for (int g = 0; g < GOUT; ++g)
        if (wacc[g] > best) { best = wacc[g]; bi = g; }
      tv[j] = best; tix[j] = bi; wacc[bi] = -INFINITY;
    }
    const float mx = tv[0];
    float ex[TOPK]; float sum = 0.f;
    for (int j = 0; j < TOPK; ++j) { ex[j] = __expf(tv[j] - mx); sum += ex[j]; }
    const float inv = 1.f / sum;
    for (int j = 0; j < TOPK; ++j) grow[tix[j]] = ex[j] * inv;
  }
  __syncthreads();
  if (tid < GOUT) gatesOut[(size_t)b * GOUT + tid] = grow[tid];
}

// ---------------------------------------------------------------------------
// Kernel 2: dispatch. Single block. Thread e collects ascending batch rows
// with gates[b,e] > 0 (== stable sort of row-major nonzero by expert column).
// Thread 0 then builds exclusive float-offsets into d_out.
// ---------------------------------------------------------------------------
__global__ void __launch_bounds__(128)
dispatch_kernel(const float* __restrict__ gates,
                int* __restrict__ counts,
                int* __restrict__ rowIdx,
                int* __restrict__ outOff) {
  const int e = threadIdx.x;
  if (e < N_EXPERTS) {
    int c = 0;
    for (int b = 0; b < B_BATCH; ++b)
      if (gates[(size_t)b * GOUT + e] > 0.f) rowIdx[e * B_BATCH + (c++)] = b;
    counts[e] = c;
  }
  __syncthreads();
  if (threadIdx.x == 0) {
    long long off = 0;
    for (int i = 0; i < N_EXPERTS; ++i) {
      outOff[i] = (int)off;
      const int p  = i + 2;
      const int nP = (SEQ_LEN + p - 1) / p;
      off += (long long)counts[i] * nP * D_MODEL;
    }
  }
}

// ---------------------------------------------------------------------------
// Kernel 3: expert patch embedding via V_WMMA_F32_16X16X4_F32.
// grid = (32 column tiles, 95 experts), 256 threads = 8 wave32s.
// B strip value_w[e][:,nt*16:+16] staged in LDS once per block using
// GLOBAL_LOAD_ASYNC_TO_LDS_B128 (ASYNCcnt) when available; value_w streams
// from HBM exactly once (~11us floor at 23.3 TB/s). K padded to a multiple
// of 16 so the K-loop runs 4 WMMAs per iteration; the unfold index k=c*p+t
// is tracked incrementally -- no integer division, no exec divergence.
// ---------------------------------------------------------------------------
__global__ void __launch_bounds__(256)
expert_kernel(const float* __restrict__ x,
              const float* __restrict__ value_w,
              const int* __restrict__ counts,
              const int* __restrict__ rowIdx,
              const int* __restrict__ outOff,
              float* __restrict__ out) {
  const int nt = blockIdx.x;          // 0..31 : 16-wide column tile of 512
  const int e  = blockIdx.y;          // expert / gate column
  const int nRows = counts[e];
  if (nRows == 0) return;

  const int p      = e + 2;
  const int nP     = (SEQ_LEN + p - 1) / p;     // patches per row
  const int K      = NUM_VARS * p;              // reduction length
  const int Kpad   = (K + 15) & ~15;            // pad K to multiple of 16
  const int mTiles = (nP + 15) >> 4;
  const int ntBase = nt * 16;

  extern __shared__ float Bs[];                 // [Kpad][16], pad rows = 0

  const float* Wp = value_w + (size_t)e * MAX_IN * D_MODEL;
#if USE_ASYNC_LDS
  // one 16B chunk per thread-iteration; 4 chunks per 16-float B row
  for (int ch = threadIdx.x; ch < K * 4; ch += blockDim.x) {
    const int k = ch >> 2, ng = (ch & 3) << 2;
    const float* g = Wp + (size_t)k * D_MODEL + ntBase + ng;
    float* l = Bs + (ch << 2);
    __builtin_amdgcn_global_load_async_to_lds_b128(
        (glob128_t)g, (lds128_t)l, 0, 0);
  }
  for (int idx = K * 16 + threadIdx.x; idx < Kpad * 16; idx += blockDim.x)
    Bs[idx] = 0.f;                              // zero pad rows
  __builtin_amdgcn_s_wait_asynccnt(0);          // own wave's async loads done
#else
  for (int idx = threadIdx.x; idx < Kpad * 16; idx += blockDim.x) {
    const int k = idx >> 4, n = idx & 15;
    float v = 0.f;
    if (k < K) {
      v = Wp[(size_t)k * D_MODEL + ntBase + n];
      __builtin_prefetch(Wp + (size_t)(k + 64) * D_MODEL + ntBase + n, 0, 1);
    }
    Bs[idx] = v;
  }
#endif
  __syncthreads();

  const int lane  = threadIdx.x & 31;
  const int wave  = __builtin_amdgcn_readfirstlane(threadIdx.x >> 5); // scalar
  const int mSub  = lane & 15;            // A: row within tile; B/D: column
  const int kOff  = (lane >> 4) << 1;     // lanes 0-15 -> K+0/1, 16-31 -> K+2/3
  const int dim   = ntBase + mSub;        // output column (model dim)
  const int kSteps = Kpad >> 2;           // multiple of 4

  // sinusoidal PE: div = exp(-(ln 1e4)/512 * (dim rounded down to even))
  const float div = __expf((-9.21034037197618f / 512.0f) * (float)(dim & ~1));
  const bool  odd = (dim & 1) != 0;

  const int items = nRows * mTiles;
  for (int item = wave; item < items; item += 8) {
    const int mt   = item % mTiles;               // scalar (mTiles <= 3)
    const int slot = item / mTiles;
    const int brow = rowIdx[e * B_BATCH + slot];  // scalar load
    const float* xb = x + (size_t)brow * SEQ_LEN * NUM_VARS;

    const int  m      = mt * 16 + mSub;   // patch index this lane feeds as A-row
    const bool mValid = m < nP;
    const int  mp     = m * p;

    // incremental unfold coordinates for k = 4*ks + kOff: k = c*p + t
    int c0 = 0, t0 = kOff;
    if (t0 >= p) { t0 -= p; c0 = 1; }     // kOff<=2, p>=2: at most one wrap

    const float* bptr = Bs + kOff * 16 + mSub;

    v8f acc = {0.f, 0.f, 0.f, 0.f, 0.f, 0.f, 0.f, 0.f};
    for (int ks = 0; ks < kSteps; ks += 4) {
      v2f aF[4], bF[4];
#pragma unroll
      for (int j = 0; j < 4; ++j) {
        // element k0 = c0*p + t0 ; element k0+1 derived with one wrap
        int t1 = t0 + 1, c1 = c0;
        if (t1 >= p) { t1 = 0; ++c1; }
        int tt0 = mp + t0; tt0 = tt0 > SEQ_LEN - 1 ? SEQ_LEN - 1 : tt0;
        int tt1 = mp + t1; tt1 = tt1 > SEQ_LEN - 1 ? SEQ_LEN - 1 : tt1;
        const int cc0 = c0 < NUM_VARS ? c0 : NUM_VARS - 1;
        const int cc1 = c1 < NUM_VARS ? c1 : NUM_VARS - 1;
        const float v0 = xb[tt0 * NUM_VARS + cc0];   // always in-bounds
        const float v1 = xb[tt1 * NUM_VARS + cc1];
        aF[j].x = (mValid && c0 < NUM_VARS) ? v0 : 0.f;
        aF[j].y = (mValid && c1 < NUM_VARS) ? v1 : 0.f;
        bF[j].x = bptr[0];
        bF[j].y = bptr[16];
        bptr += 64;
        // advance k by 4: at most two wraps since t0+4 < 3p for p >= 2
        t0 += 4;
        if (t0 >= p) { t0 -= p; ++c0; }
        if (t0 >= p) { t0 -= p; ++c0; }
      }
#pragma unroll
      for (int j = 0; j < 4; ++j) {
        acc = __builtin_amdgcn_wmma_f32_16x16x4_f32(
            /*neg_a=*/false, aF[j], /*neg_b=*/false, bF[j],
            /*c_mod=*/(short)0, acc, /*reuse_a=*/false, /*reuse_b=*/false);
      }
    }

    // epilogue: C/D layout VGPR r -> (M=r | M=r+8), add PE (branchless), store
    const long long base = (long long)outOff[e] + (long long)slot * nP * D_MODEL;
#pragma unroll
    for (int r = 0; r < 8; ++r) {
      const int ml = r + ((lane >> 4) << 3);
      const int mg = mt * 16 + ml;
      if (mg < nP) {
        float s, c;
        __sincosf((float)mg * div, &s, &c);
        out[base + (long long)mg * D_MODEL + dim] = acc[r] + (odd ? c : s);
      }
    }
  }
}

// ---------------------------------------------------------------------------
extern "C" void kernel_launch(void* const* d_in, const int* in_sizes, int n_in,
                              void* d_out, int out_size, void* d_ws, size_t ws_size,
                              hipStream_t stream) {
  const float* x       = (const float*)d_in[0];
  const float* w1      = (const float*)d_in[1];
  const float* b1      = (const float*)d_in[2];
  const float* w2      = (const float*)d_in[3];
  const float* b2      = (const float*)d_in[4];
  const float* value_w = (const float*)d_in[5];

  float* out      = (float*)d_out;
  float* gatesOut = out + ((size_t)out_size - (size_t)B_BATCH * GOUT); // tuple tail

  int* counts = (int*)d_ws;                    // [128]  (95 used)
  int* rowIdx = counts + 128;                  // [95*512]
  int* outOff = rowIdx + N_EXPERTS * B_BATCH;  // [128]  (95 used)

  gates_kernel<<<B_BATCH, 192, 0, stream>>>(x, w1, b1, w2, b2, gatesOut);
  dispatch_kernel<<<1, 128, 0, stream>>>(gatesOut, counts, rowIdx, outOff);
  // dynamic LDS sized for widest expert: Kpad_max*16 floats = 1344*16*4 bytes
  expert_kernel<<<dim3(32, N_EXPERTS), 256, MAX_IN * 16 * sizeof(float), stream>>>(
      x, value_w, counts, rowIdx, outOff, out);
}